// CompGCN_37778532335712
// MI455X (gfx1250) — compile-verified
//
#include <hip/hip_runtime.h>

#define DIM 128

typedef __attribute__((ext_vector_type(2))) float v2f;
typedef __attribute__((ext_vector_type(8))) float v8f;

// ---------------------------------------------------------------------------
// CSR-by-destination build (avoids 153M fp32 atomics in favor of ~1.2M int)
// ---------------------------------------------------------------------------
__global__ void zero_i32_kernel(int* __restrict__ p, int n) {
  int i = blockIdx.x * blockDim.x + threadIdx.x;
  if (i < n) p[i] = 0;
}

__global__ void hist_kernel(const int* __restrict__ dst, int* __restrict__ cnt, int E) {
  int e = blockIdx.x * blockDim.x + threadIdx.x;
  if (e < E) atomicAdd(&cnt[dst[e]], 1);
}

__global__ void partial_sum_kernel(const int* __restrict__ cnt, int* __restrict__ part, int n) {
  __shared__ int buf[512];
  int i = blockIdx.x * 512 + threadIdx.x;
  buf[threadIdx.x] = (i < n) ? cnt[i] : 0;
  __syncthreads();
  for (int off = 256; off > 0; off >>= 1) {
    if (threadIdx.x < off) buf[threadIdx.x] += buf[threadIdx.x + off];
    __syncthreads();
  }
  if (threadIdx.x == 0) part[blockIdx.x] = buf[0];
}

__global__ void scan_partials_kernel(int* __restrict__ part, int nb) {
  if (blockIdx.x == 0 && threadIdx.x == 0) {
    int run = 0;
    for (int b = 0; b < nb; ++b) { int t = part[b]; part[b] = run; run += t; }
  }
}

__global__ void local_scan_kernel(const int* __restrict__ cnt, const int* __restrict__ part,
                                  int* __restrict__ ptr, int n) {
  __shared__ int buf[512];
  int i = blockIdx.x * 512 + threadIdx.x;
  buf[threadIdx.x] = (i < n) ? cnt[i] : 0;
  __syncthreads();
  // inclusive Hillis-Steele scan
  for (int off = 1; off < 512; off <<= 1) {
    int t = (threadIdx.x >= off) ? buf[threadIdx.x - off] : 0;
    __syncthreads();
    buf[threadIdx.x] += t;
    __syncthreads();
  }
  if (i < n) ptr[i + 1] = part[blockIdx.x] + buf[threadIdx.x];
  if (i == 0) ptr[0] = 0;
}

__global__ void fill_kernel(const int* __restrict__ dst, const int* __restrict__ ptr,
                            int* __restrict__ cur, int* __restrict__ eid, int E) {
  int e = blockIdx.x * blockDim.x + threadIdx.x;
  if (e < E) {
    int d = dst[e];
    int pos = ptr[d] + atomicAdd(&cur[d], 1);
    eid[pos] = e;
  }
}

// ---------------------------------------------------------------------------
// h = node_emb[x]  (512B-per-row coalesced gather)
// ---------------------------------------------------------------------------
__global__ void gather_kernel(const int* __restrict__ x, const float* __restrict__ emb,
                              float* __restrict__ h, int Nn) {
  int idx = blockIdx.x * blockDim.x + threadIdx.x;
  int node = idx >> 5;
  if (node >= Nn) return;
  int c = (idx & 31) * 4;
  int s = x[node];
  *(float4*)(h + (size_t)node * DIM + c) = *(const float4*)(emb + (size_t)s * DIM + c);
}

// ---------------------------------------------------------------------------
// Pull-mode aggregation: one wave per destination node, no float atomics.
// agg[d] = sum over incoming edges e of h[src[e]] * rel[etype[e]]
// ---------------------------------------------------------------------------
__global__ __launch_bounds__(256) void aggregate_kernel(
    const float* __restrict__ h, const float* __restrict__ rel,
    const int* __restrict__ src, const int* __restrict__ etype,
    const int* __restrict__ ptr, const int* __restrict__ eid,
    float* __restrict__ agg, int Nn) {
  int wave = threadIdx.x >> 5;
  int lane = threadIdx.x & 31;
  int node = blockIdx.x * 8 + wave;
  if (node >= Nn) return;
  int beg = ptr[node], end = ptr[node + 1];
  int c = lane * 4;
  float ax = 0.f, ay = 0.f, az = 0.f, aw = 0.f;
  for (int i = beg; i < end; ++i) {
    int e = eid[i];                 // wave-uniform scalar loads
    int s = src[e], t = etype[e];
    const float4 hv = *(const float4*)(h + (size_t)s * DIM + c);
    const float4 rv = *(const float4*)(rel + (size_t)t * DIM + c);
    ax = fmaf(hv.x, rv.x, ax);
    ay = fmaf(hv.y, rv.y, ay);
    az = fmaf(hv.z, rv.z, az);
    aw = fmaf(hv.w, rv.w, aw);
  }
  float4 o; o.x = ax; o.y = ay; o.z = az; o.w = aw;
  *(float4*)(agg + (size_t)node * DIM + c) = o;
}

// ---------------------------------------------------------------------------
// out = A @ W^T + bias, optionally fused LayerNorm + ReLU.
// Block = 256 threads = 8 waves; block computes a 16-row x 128-col strip.
// Each wave: 16x16 tile via 32x V_WMMA_F32_16X16X4_F32 over K=128.
// ---------------------------------------------------------------------------
template <bool DO_LN>
__global__ __launch_bounds__(256) void gemm_ln_kernel(
    const float* __restrict__ A,     // [M, 128]
    const float* __restrict__ W,     // [128, 128] row-major; out = A @ W^T
    const float* __restrict__ bias,  // [128]
    const float* __restrict__ gamma, // [128] (DO_LN only)
    const float* __restrict__ beta,  // [128] (DO_LN only)
    float* __restrict__ out, int M) {
  __shared__ float lA[16][132];         // padded: conflict-free A-frag reads
  __shared__ float lO[16][132];
  __shared__ float red[2][16][16];

  const int tid = threadIdx.x;
  const int wave = tid >> 5;
  const int lane = tid & 31;
  const int row0 = blockIdx.x * 16;

  // cooperative load of the 16x128 A strip into LDS
  {
    int r = tid >> 4;
    int c = (tid & 15) * 8;
    if (row0 + r < M) {
      const float* s = A + (size_t)(row0 + r) * DIM + c;
#pragma unroll
      for (int j = 0; j < 8; ++j) lA[r][c + j] = s[j];
    } else {
#pragma unroll
      for (int j = 0; j < 8; ++j) lA[r][c + j] = 0.f;
    }
  }
  __syncthreads();

  // WMMA f32 16x16x4 fragment indexing (ISA 7.12.2):
  //   A: lane -> (M = lane&15, K = 2*(lane>>4) + v)   v in {0,1}
  //   B: lane -> (K = 2*(lane>>4) + v, N = lane&15)
  //   D: vgpr r -> row r + 8*(lane>>4), col = lane&15
  const int n = lane & 15;
  const int kh = lane >> 4;
  const int col0 = wave * 16;
  const float* wrow = W + (size_t)(col0 + n) * DIM + 2 * kh;  // B[k][n] = W[col0+n][k]

  v8f acc = {};
#pragma unroll
  for (int kk = 0; kk < DIM; kk += 4) {
    v2f a, b;
    a.x = lA[n][kk + 2 * kh];
    a.y = lA[n][kk + 2 * kh + 1];
    b.x = wrow[kk];
    b.y = wrow[kk + 1];
    acc = __builtin_amdgcn_wmma_f32_16x16x4_f32(
        /*neg_a=*/false, a, /*neg_b=*/false, b,
        /*c_mod=*/(short)0, acc, /*reuse_a=*/false, /*reuse_b=*/false);
  }

  // spill D tiles to LDS so the block can do the cross-column LayerNorm
  {
    int m0 = 8 * kh;
#pragma unroll
    for (int r = 0; r < 8; ++r) lO[m0 + r][col0 + n] = acc[r];
  }
  __syncthreads();

  // each thread owns row = tid/16, cols (tid&15)*8 .. +7
  int r = tid >> 4;
  int cb = (tid & 15) * 8;
  float vals[8];
  float s1 = 0.f, s2 = 0.f;
#pragma unroll
  for (int j = 0; j < 8; ++j) {
    float v = lO[r][cb + j] + bias[cb + j];
    vals[j] = v;
    s1 += v;
    s2 += v * v;
  }
  if (DO_LN) {
    red[0][r][tid & 15] = s1;
    red[1][r][tid & 15] = s2;
    __syncthreads();
    float sum = 0.f, sq = 0.f;
#pragma unroll
    for (int j = 0; j < 16; ++j) { sum += red[0][r][j]; sq += red[1][r][j]; }
    float mean = sum * (1.f / 128.f);
    float var = sq * (1.f / 128.f) - mean * mean;
    float rstd = rsqrtf(var + 1e-5f);
#pragma unroll
    for (int j = 0; j < 8; ++j) {
      float v = (vals[j] - mean) * rstd * gamma[cb + j] + beta[cb + j];
      vals[j] = v > 0.f ? v : 0.f;
    }
  }
  if (row0 + r < M) {
    float* dp = out + (size_t)(row0 + r) * DIM + cb;
#pragma unroll
    for (int j = 0; j < 8; ++j) dp[j] = vals[j];
  }
}

// ---------------------------------------------------------------------------
extern "C" void kernel_launch(void* const* d_in, const int* in_sizes, int n_in,
                              void* d_out, int out_size, void* d_ws, size_t ws_size,
                              hipStream_t stream) {
  const int Nn = in_sizes[0];       // 50000
  const int E = in_sizes[2];        // 600000

  const int* x = (const int*)d_in[0];
  const int* ei = (const int*)d_in[1];
  const int* et = (const int*)d_in[2];
  const float* node_emb = (const float*)d_in[3];
  const float* rel1 = (const float*)d_in[4];
  // d_in[5]/d_in[6] (rel_w1/rel_b1): layer-1 updated_rels are discarded
  const float* nw1 = (const float*)d_in[7];
  const float* nb1 = (const float*)d_in[8];
  const float* g1 = (const float*)d_in[9];
  const float* b1 = (const float*)d_in[10];
  const float* rel2 = (const float*)d_in[11];
  const float* rw2 = (const float*)d_in[12];
  const float* rb2 = (const float*)d_in[13];
  const float* nw2 = (const float*)d_in[14];
  const float* nb2 = (const float*)d_in[15];
  const float* g2 = (const float*)d_in[16];
  const float* b2 = (const float*)d_in[17];

  const int* srcv = ei;
  const int* dstv = ei + E;

  float* h = (float*)d_out;                        // [Nn, 128]
  float* rout = (float*)d_out + (size_t)Nn * DIM;  // [400, 128]

  // workspace layout (~28.4 MB)
  char* ws = (char*)d_ws;
  float* agg = (float*)ws;  ws += (size_t)Nn * DIM * sizeof(float);
  int* ptr = (int*)ws;      ws += (size_t)(Nn + 1) * sizeof(int);
  int* cnt = (int*)ws;      ws += (size_t)Nn * sizeof(int);
  int* eid = (int*)ws;      ws += (size_t)E * sizeof(int);
  const int nb_scan = (Nn + 511) / 512;
  int* part = (int*)ws;     ws += (size_t)nb_scan * sizeof(int);

  // --- build CSR-by-destination (rebuilt deterministically every call) ---
  zero_i32_kernel<<<(Nn + 255) / 256, 256, 0, stream>>>(cnt, Nn);
  hist_kernel<<<(E + 255) / 256, 256, 0, stream>>>(dstv, cnt, E);
  partial_sum_kernel<<<nb_scan, 512, 0, stream>>>(cnt, part, Nn);
  scan_partials_kernel<<<1, 64, 0, stream>>>(part, nb_scan);
  local_scan_kernel<<<nb_scan, 512, 0, stream>>>(cnt, part, ptr, Nn);
  zero_i32_kernel<<<(Nn + 255) / 256, 256, 0, stream>>>(cnt, Nn);
  fill_kernel<<<(E + 255) / 256, 256, 0, stream>>>(dstv, ptr, cnt, eid, E);

  // --- h = node_emb[x] ---
  gather_kernel<<<((size_t)Nn * 32 + 255) / 256, 256, 0, stream>>>(x, node_emb, h, Nn);

  // --- layer 1: aggregate -> GEMM+bias+LN+ReLU ---
  aggregate_kernel<<<(Nn + 7) / 8, 256, 0, stream>>>(h, rel1, srcv, et, ptr, eid, agg, Nn);
  gemm_ln_kernel<true><<<(Nn + 15) / 16, 256, 0, stream>>>(agg, nw1, nb1, g1, b1, h, Nn);

  // --- layer 2 ---
  aggregate_kernel<<<(Nn + 7) / 8, 256, 0, stream>>>(h, rel2, srcv, et, ptr, eid, agg, Nn);
  gemm_ln_kernel<true><<<(Nn + 15) / 16, 256, 0, stream>>>(agg, nw2, nb2, g2, b2, h, Nn);

  // --- rels = rel_emb2 @ rel_w2^T + rel_b2 (no LN/ReLU) ---
  const int R = in_sizes[11] / DIM;  // 400
  gemm_ln_kernel<false><<<(R + 15) / 16, 256, 0, stream>>>(rel2, rw2, rb2, nullptr, nullptr, rout, R);
}